// RendererMonoNPHM_52286931861556
// MI455X (gfx1250) — compile-verified
//
#include <hip/hip_runtime.h>
#include <math.h>

// ---------------------------------------------------------------------------
// RendererMonoNPHM for gfx1250 (MI455X).
// Bandwidth-bound streaming + per-ray scan workload: no matmul structure, so
// no WMMA; the CDNA5 features used are the Tensor Data Mover in BOTH
// directions:
//   - tensor_load_to_lds:    t_rand tile -> LDS with hardware row padding
//                            (stride 65 dwords -> bank-conflict-free rows)
//   - tensor_store_from_lds: packed weights tile (row stride 63 == -1 mod 64,
//                            also conflict-free) -> contiguous global output
// both synchronized with s_wait_tensorcnt.
// ---------------------------------------------------------------------------

#define ROWS 64            // rays per block (2 waves of 32)
#define LDS_STRIDE 65      // 64 payload dwords + 1 pad dword (TDM pad_amount=1)
#define WSTRIDE 63         // packed weights row stride (conflict-free rotation)

typedef __attribute__((ext_vector_type(4))) unsigned int u32x4;
typedef __attribute__((ext_vector_type(8))) int i32x8;
typedef __attribute__((ext_vector_type(4))) int i32x4;

#if defined(__HIP_DEVICE_COMPILE__) && __has_builtin(__builtin_amdgcn_tensor_load_to_lds)
#define HAVE_TDM_LD 1
#else
#define HAVE_TDM_LD 0
#endif
#if defined(__HIP_DEVICE_COMPILE__) && __has_builtin(__builtin_amdgcn_tensor_store_from_lds)
#define HAVE_TDM_ST 1
#else
#define HAVE_TDM_ST 0
#endif

#if HAVE_TDM_LD || HAVE_TDM_ST
__device__ __forceinline__ void tdm_wait0() {
#if __has_builtin(__builtin_amdgcn_s_wait_tensorcnt)
    __builtin_amdgcn_s_wait_tensorcnt(0);
#else
    asm volatile("s_wait_tensorcnt 0" ::: "memory");
#endif
}
#endif

// ---------------------------------------------------------------------------
__global__ void init_ws_kernel(float* ws) {
    ws[0] = 0.0f;                  // sum si0 (masked)
    ws[1] = 0.0f;                  // sum si1 (masked)
    ((unsigned*)ws)[2] = 0u;       // hit count
}

// ---------------------------------------------------------------------------
// Pass 1: global reduction of masked sphere-intersection depths.
__global__ __launch_bounds__(256) void reduce_kernel(
    const float* __restrict__ rd, const float* __restrict__ cam,
    float* __restrict__ ws, int n)
{
    int i = blockIdx.x * blockDim.x + threadIdx.x;
    float s0 = 0.0f, s1 = 0.0f;
    int cnt = 0;
    if (i < n) {
        float dx = rd[3 * i + 0], dy = rd[3 * i + 1], dz = rd[3 * i + 2];
        float cx = cam[0], cy = cam[1], cz = cam[2];
        float dotc = dx * cx + dy * cy + dz * cz;
        float cam2 = cx * cx + cy * cy + cz * cz;
        float under = dotc * dotc - (cam2 - 1.0f);   // R_SPHERE = 1
        if (under > 0.0f) {
            float sq = sqrtf(under);
            s0 = -sq - dotc;
            s1 =  sq - dotc;
            cnt = 1;
        }
    }
    // wave32 reduction
    for (int off = 16; off > 0; off >>= 1) {
        s0  += __shfl_down(s0, off);
        s1  += __shfl_down(s1, off);
        cnt += __shfl_down(cnt, off);
    }
    if ((threadIdx.x & 31) == 0) {
        atomicAdd(&ws[0], s0);
        atomicAdd(&ws[1], s1);
        atomicAdd(((unsigned*)ws) + 2, (unsigned)cnt);
    }
}

// ---------------------------------------------------------------------------
// Pass 2: per-ray pipeline. One thread = one ray.
__global__ __launch_bounds__(ROWS) void main_kernel(
    const float* __restrict__ rd, const float* __restrict__ cam,
    const float* __restrict__ trand, const float* __restrict__ ws,
    float* __restrict__ ns, float* __restrict__ zs, float* __restrict__ wt,
    int n)
{
    __shared__ float sSteps[ROWS * LDS_STRIDE];  // t_rand tile, then step depths
    __shared__ float sW[ROWS * WSTRIDE];         // packed weights, then CDF

    const int tid  = threadIdx.x;
    const int ray0 = blockIdx.x * ROWS;
    const int ray  = ray0 + tid;
    const unsigned rem =
        (unsigned)((n - ray0) < ROWS ? (n - ray0) : ROWS);

#if HAVE_TDM_LD
    if (tid < 32) {   // one wave issues the DMA (EXEC ignored by TDM)
        unsigned lds_off = (unsigned)(unsigned long long)(void*)sSteps;
        unsigned long long ga =
            (unsigned long long)(const void*)(trand + (size_t)ray0 * 64);

        // D# group0: count=1 | lds_addr | global_addr[56:0] | type=2
        u32x4 g0;
        g0.x = 1u;
        g0.y = lds_off;
        g0.z = (unsigned)ga;
        g0.w = (unsigned)((ga >> 32) & 0x01FFFFFFull) | 0x80000000u;

        // D# group1: data_size=4B, pad_enable, pad_interval=64dw, pad_amount=1dw,
        // tensor_dim0=64, tensor_dim1=rem, tile_dim0=64, tile_dim1=ROWS,
        // tensor_dim0_stride=64 elements.
        i32x8 g1 = {
            (int)((2u << 16) | (1u << 20) | (5u << 22)),  // dword0
            (int)(64u << 16),                              // tensor_dim0[15:0]
            (int)(rem << 16),                              // tensor_dim1[15:0]
            (int)(64u << 16),                              // tile_dim0
            (int)ROWS,                                     // tile_dim1
            64,                                            // tensor_dim0_stride
            0, 0 };

        i32x4 gz = {0, 0, 0, 0};
#if defined(__clang_major__) && __clang_major__ >= 23
        i32x8 z8 = {0, 0, 0, 0, 0, 0, 0, 0};
        __builtin_amdgcn_tensor_load_to_lds(g0, g1, gz, gz, z8, 0);
#else
        __builtin_amdgcn_tensor_load_to_lds(g0, g1, gz, gz, 0);
#endif
        tdm_wait0();
    }
#else
    // Fallback: coalesced copy global -> padded LDS.
    for (int e = tid; e < ROWS * 64; e += ROWS) {
        int r = e >> 6, c = e & 63;
        if (ray0 + r < n)
            sSteps[r * LDS_STRIDE + c] = trand[(size_t)(ray0 + r) * 64 + c];
    }
#endif
    __syncthreads();

    const float cx = cam[0], cy = cam[1], cz = cam[2];
    const float sum0 = ws[0], sum1 = ws[1];
    const unsigned hitc = ((const unsigned*)ws)[2];
    const float nh = (float)(hitc ? hitc : 1u);
    const float m0 = sum0 / nh, m1 = sum1 / nh;

    const bool act = (ray < n);
    float dx = 0.f, dy = 0.f, dz = 0.f;
    float wsum = 1.0f;
    float* row  = &sSteps[tid * LDS_STRIDE];
    float* wrow = &sW[tid * WSTRIDE];

    if (act) {
        dx = rd[3 * ray + 0]; dy = rd[3 * ray + 1]; dz = rd[3 * ray + 2];
        float dotc = dx * cx + dy * cy + dz * cz;
        float cam2 = cx * cx + cy * cy + cz * cz;
        float under = dotc * dotc - (cam2 - 1.0f);
        bool hit = under > 0.0f;
        float sq = sqrtf(hit ? under : 0.0f);
        float si0 = hit ? (-sq - dotc) : m0;
        float si1 = hit ? ( sq - dotc) : m1;
        float mind = fmaxf(si0, 0.0f);
        float maxd = fmaxf(si1, 0.0f);
        float rng = maxd - mind;
        float sdist = rng * (1.0f / 64.0f);
        float lw    = rng * (1.0f / 63.0f);

        const float invs = 20.085537f;           // exp(0.3*10)
        float prev = 0.0f, trans = 1.0f;
        wsum = 0.0f;
        for (int i = 0; i < 64; ++i) {
            float t = fmaf(lw, (float)i, mind) + (row[i] - 0.5f) * sdist;
            row[i] = t;                           // overwrite t_rand with step depth
            float px = fmaf(t, dx, cx);
            float py = fmaf(t, dy, cy);
            float pz = fmaf(t, dz, cz);
            float sdf = sqrtf(px * px + py * py + pz * pz) - 0.5f;
            float csig = 1.0f / (1.0f + expf(-invs * sdf));
            if (i > 0) {
                float alpha = (prev - csig + 1e-8f) / (prev + 1e-8f);
                alpha = fminf(fmaxf(alpha, 0.0f), 1.0f);
                float w = alpha * trans;
                wrow[i - 1] = w;
                wsum += w + 1e-8f;
                trans *= (1.0f - alpha + 1e-7f);
            }
            prev = csig;
        }
    }
    __syncthreads();   // all weight rows complete in LDS

#if HAVE_TDM_ST
    // DMA the packed weights tile straight to the contiguous global output.
    // 2-D tile: dim0 = 63 dwords/row, dim1 = ROWS rows; tensor_dim1 = rem so
    // tail rows past the grid end are dropped by TDM OOB semantics.
    if (tid < 32) {
        unsigned lds_off = (unsigned)(unsigned long long)(void*)sW;
        unsigned long long ga =
            (unsigned long long)(void*)(wt + (size_t)ray0 * 63);

        u32x4 g0;
        g0.x = 1u;
        g0.y = lds_off;
        g0.z = (unsigned)ga;
        g0.w = (unsigned)((ga >> 32) & 0x01FFFFFFull) | 0x80000000u;

        i32x8 g1 = {
            (int)(2u << 16),                               // data_size=4B, no pad
            (int)(63u << 16),                              // tensor_dim0 = 63
            (int)(rem << 16),                              // tensor_dim1 = rem
            (int)(63u << 16),                              // tile_dim0 = 63
            (int)ROWS,                                     // tile_dim1 = ROWS
            63,                                            // tensor_dim0_stride
            0, 0 };

        i32x4 gz = {0, 0, 0, 0};
#if defined(__clang_major__) && __clang_major__ >= 23
        i32x8 z8 = {0, 0, 0, 0, 0, 0, 0, 0};
        __builtin_amdgcn_tensor_store_from_lds(g0, g1, gz, gz, z8, 0);
#else
        __builtin_amdgcn_tensor_store_from_lds(g0, g1, gz, gz, 0);
#endif
        tdm_wait0();   // TDM must finish reading LDS before in-place CDF overwrite
    }
#else
    {
        float* wbase = wt + (size_t)ray0 * 63;
        const int tot = (int)rem * 63;
        for (int e = tid; e < tot; e += ROWS) wbase[e] = sW[e];
    }
#endif
    __syncthreads();

    if (act) {
        // In-place weights -> CDF (len 64, cdf[0]=0), matching reference order.
        float inv_sum = 1.0f / wsum;
        float acc = 0.0f;
        float wprev = wrow[0];
        wrow[0] = 0.0f;
        for (int k = 1; k < 64; ++k) {
            float wn = (k < 63) ? wrow[k] : 0.0f;
            acc += (wprev + 1e-8f) * inv_sum;
            wrow[k] = acc;
            wprev = wn;
        }

        // Merge-scan searchsorted (u ascending, cdf ascending) + lerp.
        float* zrow = zs + (size_t)ray * 32;
        float* nrow = ns + (size_t)ray * 96;
        int k = 1;
        for (int j = 0; j < 32; ++j) {
            float u = ((float)j + 0.5f) * 0.03125f;   // (j+0.5)/32
            while (k < 64 && wrow[k] <= u) ++k;
            int b = k - 1;
            int a = (k < 64) ? k : 63;
            float cb = wrow[b], ca = wrow[a];
            float den = ca - cb;
            den = (den < 1e-8f) ? 1.0f : den;
            float tt = (u - cb) / den;
            float sb = row[b], sa = row[a];
            float z = fmaf(tt, sa - sb, sb);
            zrow[j] = z;
            nrow[3 * j + 0] = fmaf(z, dx, cx);
            nrow[3 * j + 1] = fmaf(z, dy, cy);
            nrow[3 * j + 2] = fmaf(z, dz, cz);
        }
    }
}

// ---------------------------------------------------------------------------
extern "C" void kernel_launch(void* const* d_in, const int* in_sizes, int n_in,
                              void* d_out, int out_size, void* d_ws, size_t ws_size,
                              hipStream_t stream) {
    const float* rd   = (const float*)d_in[0];   // (1, P, 3)
    const float* cam  = (const float*)d_in[1];   // (1, 3)
    const float* trnd = (const float*)d_in[2];   // (P, 64)
    const int nrays = in_sizes[0] / 3;

    float* out = (float*)d_out;
    float* ns = out;                               // (P, 32, 3)
    float* zs = out + (size_t)nrays * 96;          // (P, 32)
    float* wt = zs  + (size_t)nrays * 32;          // (P, 63)

    float* wsf = (float*)d_ws;

    init_ws_kernel<<<1, 1, 0, stream>>>(wsf);

    int rb = (nrays + 255) / 256;
    reduce_kernel<<<rb, 256, 0, stream>>>(rd, cam, wsf, nrays);

    int mb = (nrays + ROWS - 1) / ROWS;
    main_kernel<<<mb, ROWS, 0, stream>>>(rd, cam, trnd, wsf, ns, zs, wt, nrays);
}